// SGC_4569845203315
// MI455X (gfx1250) — compile-verified
//
#include <hip/hip_runtime.h>
#include <math.h>

typedef float v2f __attribute__((ext_vector_type(2)));
typedef float v8f __attribute__((ext_vector_type(8)));

#define FEAT   64
#define NCLS   40
#define NCLSP  48   // classes padded to 3x16 WMMA tiles

// ---------------- utility kernels ----------------

__global__ void k_fill(float* __restrict__ p, float val, int n) {
    int i = blockIdx.x * blockDim.x + threadIdx.x;
    if (i < n) p[i] = val;
}

// deg[col[e]] += 1  (deg pre-filled with 1.0 for the self loop)
__global__ void k_degree(const int* __restrict__ col, float* __restrict__ deg, int E) {
    int i = blockIdx.x * blockDim.x + threadIdx.x;
    if (i < E) atomicAdd(&deg[col[i]], 1.0f);
}

// deg -> dinv in place
__global__ void k_rsqrt_inplace(float* __restrict__ d, int n) {
    int i = blockIdx.x * blockDim.x + threadIdx.x;
    if (i < n) {
        float v = d[i];
        d[i] = (v > 0.0f) ? rsqrtf(v) : 0.0f;
    }
}

// xout[i] = dinv[node]^2 * xin[i]   (self-loop term seeds the accumulator)
__global__ void k_selfloop_init(const float* __restrict__ xin, float* __restrict__ xout,
                                const float* __restrict__ dinv, int total) {
    int i = blockIdx.x * blockDim.x + threadIdx.x;
    if (i < total) {
        float w = dinv[i >> 6];   // FEAT == 64
        xout[i] = w * w * xin[i];
    }
}

// edge-parallel SpMM: 16 lanes per edge, float4 gather + 4x atomic f32 scatter.
// x fits in the 192MB L2, so gathers/atomics are L2-resident.
__global__ void k_edge_spmm(const float* __restrict__ xin, float* __restrict__ xout,
                            const int* __restrict__ row, const int* __restrict__ col,
                            const float* __restrict__ dinv, int E) {
    long long t = (long long)blockIdx.x * blockDim.x + threadIdx.x;
    int e    = (int)(t >> 4);
    int lane = (int)(t & 15);
    if (e >= E) return;
    int r = row[e];
    int c = col[e];
    float w = dinv[r] * dinv[c];
    const float4 v = ((const float4*)(xin + (size_t)r * FEAT))[lane];
    float* dst = xout + (size_t)c * FEAT + lane * 4;
    atomicAdd(dst + 0, w * v.x);
    atomicAdd(dst + 1, w * v.y);
    atomicAdd(dst + 2, w * v.z);
    atomicAdd(dst + 3, w * v.w);
}

// pad W [40x64] -> Wp [48x64] (rows 40..47 zero) so B tiles are dense
__global__ void k_pad_w(const float* __restrict__ W, float* __restrict__ Wp) {
    int i = blockIdx.x * blockDim.x + threadIdx.x;
    if (i < NCLSP * FEAT) {
        int r = i / FEAT;
        Wp[i] = (r < NCLS) ? W[i] : 0.0f;
    }
}

// ---------------- WMMA GEMM: logits = x @ W^T + b ----------------
// One wave per 16-node tile; 3 class-tiles (48 padded cols), K=64 via 16
// V_WMMA_F32_16X16X4_F32 steps per class tile.
__global__ void k_gemm_wmma(const float* __restrict__ x, const float* __restrict__ Wp,
                            const float* __restrict__ bias, float* __restrict__ out, int N) {
    const int wave   = blockIdx.x * (blockDim.x >> 5) + (threadIdx.x >> 5);
    const int lane   = threadIdx.x & 31;
    const int ntiles = (N + 15) >> 4;
    if (wave >= ntiles) return;          // wave-uniform: EXEC stays all-ones

    const int base = wave << 4;
    const int half = lane >> 4;          // 0: K 0..1, 1: K 2..3 (per 4-wide step)
    const int l16  = lane & 15;

    int m = base + l16;                  // A-row this lane feeds (N % 16 == 0 typically)
    if (m >= N) m = N - 1;
    const float* arow = x + (size_t)m * FEAT;
    const float* b0r  = Wp + (size_t)(l16)      * FEAT;
    const float* b1r  = Wp + (size_t)(16 + l16) * FEAT;
    const float* b2r  = Wp + (size_t)(32 + l16) * FEAT;

    v8f acc0 = {};
    v8f acc1 = {};
    v8f acc2 = {};

    for (int k0 = 0; k0 < FEAT; k0 += 4) {
        const int ko = k0 + 2 * half;
        v2f a  = *(const v2f*)(arow + ko);
        v2f b0 = *(const v2f*)(b0r + ko);
        v2f b1 = *(const v2f*)(b1r + ko);
        v2f b2 = *(const v2f*)(b2r + ko);
        acc0 = __builtin_amdgcn_wmma_f32_16x16x4_f32(false, a, false, b0, (short)0, acc0, false, false);
        acc1 = __builtin_amdgcn_wmma_f32_16x16x4_f32(false, a, false, b1, (short)0, acc1, false, false);
        acc2 = __builtin_amdgcn_wmma_f32_16x16x4_f32(false, a, false, b2, (short)0, acc2, false, false);
    }

    // C/D layout: VGPR j -> row (j + 8*half), col l16 within each 16-wide tile
    const int c0 = l16;
    const int c1 = 16 + l16;
    const int c2 = 32 + l16;
    const float bb0 = bias[c0];
    const float bb1 = bias[c1];
    const float bb2 = (c2 < NCLS) ? bias[c2] : 0.0f;
#pragma unroll
    for (int j = 0; j < 8; ++j) {
        const int node = base + j + 8 * half;
        if (node < N) {
            float* o = out + (size_t)node * NCLS;
            o[c0] = acc0[j] + bb0;
            o[c1] = acc1[j] + bb1;
            if (c2 < NCLS) o[c2] = acc2[j] + bb2;
        }
    }
}

// ---------------- in-place row log_softmax over 40 classes ----------------
__global__ void k_logsoftmax(float* __restrict__ out, int N) {
    int n = blockIdx.x * blockDim.x + threadIdx.x;
    if (n >= N) return;
    float* p = out + (size_t)n * NCLS;
    float v[NCLS];
    float m = -INFINITY;
#pragma unroll
    for (int c = 0; c < NCLS; ++c) { v[c] = p[c]; m = fmaxf(m, v[c]); }
    float s = 0.0f;
#pragma unroll
    for (int c = 0; c < NCLS; ++c) s += expf(v[c] - m);
    const float ls = logf(s);
#pragma unroll
    for (int c = 0; c < NCLS; ++c) p[c] = v[c] - m - ls;
}

// ---------------- driver ----------------

static inline size_t align256f(size_t nfloats) { return (nfloats + 63) & ~(size_t)63; }

extern "C" void kernel_launch(void* const* d_in, const int* in_sizes, int n_in,
                              void* d_out, int out_size, void* d_ws, size_t ws_size,
                              hipStream_t stream) {
    const float* x  = (const float*)d_in[0];
    const int*   ei = (const int*)d_in[1];
    const float* W  = (const float*)d_in[2];
    const float* bi = (const float*)d_in[3];

    const int N = in_sizes[0] / FEAT;
    const int E = in_sizes[1] / 2;
    const int* row = ei;        // source
    const int* col = ei + E;    // target / aggregation index
    const int total = N * FEAT;

    float* ws = (float*)d_ws;
    size_t off = 0;
    float* dinv = ws + off; off += align256f((size_t)N);
    float* Wp   = ws + off; off += align256f((size_t)NCLSP * FEAT);
    float* x0   = ws + off; off += align256f((size_t)total);
    float* x1   = ws + off;

    const int T = 256;
    // normalization
    k_fill<<<(N + T - 1) / T, T, 0, stream>>>(dinv, 1.0f, N);           // self-loop degree
    k_degree<<<(E + T - 1) / T, T, 0, stream>>>(col, dinv, E);
    k_rsqrt_inplace<<<(N + T - 1) / T, T, 0, stream>>>(dinv, N);
    k_pad_w<<<(NCLSP * FEAT + T - 1) / T, T, 0, stream>>>(W, Wp);

    const long long ethreads = (long long)E * 16;
    const int eblocks = (int)((ethreads + T - 1) / T);

    // hop 1: x -> x0
    k_selfloop_init<<<(total + T - 1) / T, T, 0, stream>>>(x, x0, dinv, total);
    k_edge_spmm<<<eblocks, T, 0, stream>>>(x, x0, row, col, dinv, E);
    // hop 2: x0 -> x1
    k_selfloop_init<<<(total + T - 1) / T, T, 0, stream>>>(x0, x1, dinv, total);
    k_edge_spmm<<<eblocks, T, 0, stream>>>(x0, x1, row, col, dinv, E);

    // dense head: WMMA f32 GEMM + bias, then in-place log_softmax
    const int ntiles = (N + 15) >> 4;
    const int gblocks = (ntiles + 7) / 8;        // 8 waves per 256-thread block
    k_gemm_wmma<<<gblocks, 256, 0, stream>>>(x1, Wp, bi, (float*)d_out, N);
    k_logsoftmax<<<(N + T - 1) / T, T, 0, stream>>>((float*)d_out, N);
}